// GCNBaseline_790273982720
// MI455X (gfx1250) — compile-verified
//
#include <hip/hip_runtime.h>
#include <hip/hip_bf16.h>

typedef __attribute__((ext_vector_type(2))) float v2f;
typedef __attribute__((ext_vector_type(8))) float v8f;

// ---------------------------------------------------------------------------
// Degree / normalization: deg starts at 1.0 (self loop), scatter-add edges,
// then dinv = rsqrt(deg).
// ---------------------------------------------------------------------------
__global__ void deg_init_kernel(float* __restrict__ deg, int n) {
  int i = blockIdx.x * blockDim.x + threadIdx.x;
  if (i < n) deg[i] = 1.0f;
}

__global__ void deg_accum_kernel(const int* __restrict__ dst,
                                 float* __restrict__ deg, int e) {
  int i = blockIdx.x * blockDim.x + threadIdx.x;
  if (i < e) unsafeAtomicAdd(&deg[dst[i]], 1.0f);
}

__global__ void deg_to_dinv_kernel(float* __restrict__ deg, int n) {
  int i = blockIdx.x * blockDim.x + threadIdx.x;
  if (i < n) deg[i] = rsqrtf(deg[i]);   // deg >= 1 always (self loop)
}

// ---------------------------------------------------------------------------
// WMMA fp32 GEMM: H[n_rows x 64] = X[n_rows x K] @ W[K x 64]
// W is staged into LDS once per block (dynamic shared mem = K*64*4 bytes),
// so the inner loop feeds V_WMMA_F32_16X16X4_F32 from ds_load instead of
// re-reading W from vmem in every wave.
//
// A 16x4 layout (ISA 7.12.2): lanes 0-15 hold rows M=0..15 with K={0,1} in
// VGPR0/1; lanes 16-31 hold K={2,3}.  B 4x16: VGPR v holds rows K=v (lanes
// 0-15) and K=v+2 (lanes 16-31), col = lane&15 (mirrors C/D row-halving).
// C/D 16x16: VGPR v -> row m0+v (lanes 0-15) / m0+v+8 (lanes 16-31).
// ---------------------------------------------------------------------------
__global__ __launch_bounds__(256) void gemm64_wmma_f32_kernel(
    const float* __restrict__ X, const float* __restrict__ W,
    float* __restrict__ H, int n_rows, int K) {
  extern __shared__ float sW[];        // K x 64 floats

  // Cooperative W load (all threads, BEFORE any wave-level early return).
  {
    const float4* Wv = (const float4*)W;
    float4* sWv = (float4*)sW;
    int total = K * 16;                // float4 count
    for (int i = threadIdx.x; i < total; i += blockDim.x) sWv[i] = Wv[i];
  }
  __syncthreads();

  int wave = (int)((blockIdx.x * blockDim.x + threadIdx.x) >> 5);
  int lane = threadIdx.x & 31;
  int m0 = wave * 16;
  if (m0 >= n_rows) return;            // wave-uniform: EXEC stays all-ones

  const int row  = m0 + (lane & 15);
  const int kh   = (lane >> 4) * 2;    // 0 (lanes 0-15) or 2 (lanes 16-31)
  const int ncol = lane & 15;

  v8f acc0 = {}, acc1 = {}, acc2 = {}, acc3 = {};
  const float* xrow = X + (long long)row * K;

  for (int k = 0; k < K; k += 4) {
    v2f a;
    a.x = xrow[k + kh];
    a.y = xrow[k + kh + 1];
    const float* w0 = sW + (k + kh) * 64;                // row k+kh
    const float* w1 = w0 + 64;                           // row k+kh+1
    v2f b0, b1, b2, b3;
    b0.x = w0[ncol];      b0.y = w1[ncol];
    b1.x = w0[ncol + 16]; b1.y = w1[ncol + 16];
    b2.x = w0[ncol + 32]; b2.y = w1[ncol + 32];
    b3.x = w0[ncol + 48]; b3.y = w1[ncol + 48];
    acc0 = __builtin_amdgcn_wmma_f32_16x16x4_f32(false, a, false, b0, (short)0, acc0, false, false);
    acc1 = __builtin_amdgcn_wmma_f32_16x16x4_f32(false, a, false, b1, (short)0, acc1, false, false);
    acc2 = __builtin_amdgcn_wmma_f32_16x16x4_f32(false, a, false, b2, (short)0, acc2, false, false);
    acc3 = __builtin_amdgcn_wmma_f32_16x16x4_f32(false, a, false, b3, (short)0, acc3, false, false);
  }

  int srow = m0 + (lane >> 4) * 8;     // rows v / v+8 split across lane halves
  float* out = H + ncol;
#pragma unroll
  for (int v = 0; v < 8; ++v) {
    long long base = (long long)(srow + v) * 64;
    out[base +  0] = acc0[v];
    out[base + 16] = acc1[v];
    out[base + 32] = acc2[v];
    out[base + 48] = acc3[v];
  }
}

// ---------------------------------------------------------------------------
// agg[n][c] = h[n][c] * dinv[n]^2   (self-loop contribution, float4 vector)
// ---------------------------------------------------------------------------
__global__ void self_loop_init_kernel(const float* __restrict__ h,
                                      const float* __restrict__ dinv,
                                      float* __restrict__ agg, int n) {
  long long t = (long long)blockIdx.x * blockDim.x + threadIdx.x;
  int node = (int)(t >> 4);
  if (node >= n) return;
  int c = (int)(t & 15) * 4;
  float w = dinv[node];
  w *= w;
  float4 v = *(const float4*)(h + (long long)node * 64 + c);
  v.x *= w; v.y *= w; v.z *= w; v.w *= w;
  *(float4*)(agg + (long long)node * 64 + c) = v;
}

// ---------------------------------------------------------------------------
// Edge aggregation: one wave per edge, each lane handles a float2 channel
// pair. Coalesced 256B gather of h[src], native float atomic scatter into
// agg[dst] (L2-resident: 25.6MB feature matrix << 192MB L2).
// ---------------------------------------------------------------------------
__global__ void edge_agg_kernel(const int* __restrict__ src,
                                const int* __restrict__ dst,
                                const float* __restrict__ dinv,
                                const float* __restrict__ h,
                                float* __restrict__ agg, int e) {
  long long t = (long long)blockIdx.x * blockDim.x + threadIdx.x;
  long long edge = t >> 5;
  if (edge >= e) return;
  int c = (int)(t & 31) * 2;
  int s = src[edge];
  int d = dst[edge];
  float norm = dinv[s] * dinv[d];
  float2 v = *(const float2*)(h + (long long)s * 64 + c);
  float* ap = agg + (long long)d * 64 + c;
  unsafeAtomicAdd(ap,     v.x * norm);
  unsafeAtomicAdd(ap + 1, v.y * norm);
}

// ---------------------------------------------------------------------------
// agg[n][c] = max(agg[n][c] + b[c], 0)   (float4 vector)
// ---------------------------------------------------------------------------
__global__ void bias_relu_kernel(float* __restrict__ agg,
                                 const float* __restrict__ b, int n) {
  long long t = (long long)blockIdx.x * blockDim.x + threadIdx.x;
  int node = (int)(t >> 4);
  if (node >= n) return;
  int c = (int)(t & 15) * 4;
  float4 v = *(float4*)(agg + (long long)node * 64 + c);
  v.x = fmaxf(v.x + b[c + 0], 0.0f);
  v.y = fmaxf(v.y + b[c + 1], 0.0f);
  v.z = fmaxf(v.z + b[c + 2], 0.0f);
  v.w = fmaxf(v.w + b[c + 3], 0.0f);
  *(float4*)(agg + (long long)node * 64 + c) = v;
}

// ---------------------------------------------------------------------------
// MLP head: out = relu(h @ Wh1 + bh1) @ Wh2 + bh2, weights staged in LDS.
// One node per thread; h row held in registers.
// ---------------------------------------------------------------------------
__global__ __launch_bounds__(128) void head_kernel(
    const float* __restrict__ h, const float* __restrict__ Wh1,
    const float* __restrict__ bh1, const float* __restrict__ Wh2,
    const float* __restrict__ bh2, float* __restrict__ out, int n) {
  __shared__ float sW1[64 * 32];
  __shared__ float sb1[32];
  __shared__ float sW2[64];
  __shared__ float sb2[2];
  for (int i = threadIdx.x; i < 64 * 32; i += blockDim.x) sW1[i] = Wh1[i];
  if (threadIdx.x < 32) sb1[threadIdx.x] = bh1[threadIdx.x];
  if (threadIdx.x < 64) sW2[threadIdx.x] = Wh2[threadIdx.x];
  if (threadIdx.x < 2)  sb2[threadIdx.x] = bh2[threadIdx.x];
  __syncthreads();

  int node = blockIdx.x * blockDim.x + threadIdx.x;
  if (node >= n) return;

  float hr[64];
  const float4* hp = (const float4*)(h + (long long)node * 64);
#pragma unroll
  for (int i = 0; i < 16; ++i) {
    float4 v = hp[i];
    hr[4 * i + 0] = v.x; hr[4 * i + 1] = v.y;
    hr[4 * i + 2] = v.z; hr[4 * i + 3] = v.w;
  }

  float o0 = sb2[0], o1 = sb2[1];
  for (int j = 0; j < 32; ++j) {
    float s = sb1[j];
#pragma unroll
    for (int k = 0; k < 64; ++k) s = fmaf(hr[k], sW1[k * 32 + j], s);
    s = fmaxf(s, 0.0f);
    o0 = fmaf(s, sW2[2 * j + 0], o0);
    o1 = fmaf(s, sW2[2 * j + 1], o1);
  }
  out[(long long)node * 2 + 0] = o0;
  out[(long long)node * 2 + 1] = o1;
}

// ---------------------------------------------------------------------------
// Launch
// ---------------------------------------------------------------------------
extern "C" void kernel_launch(void* const* d_in, const int* in_sizes, int n_in,
                              void* d_out, int out_size, void* d_ws, size_t ws_size,
                              hipStream_t stream) {
  const float* x   = (const float*)d_in[0];
  const int*   ei  = (const int*)d_in[1];     // [2, E] flat: row0=src, row1=dst
  const float* W1  = (const float*)d_in[2];
  const float* b1  = (const float*)d_in[3];
  const float* W2  = (const float*)d_in[4];
  const float* b2  = (const float*)d_in[5];
  const float* Wh1 = (const float*)d_in[6];
  const float* bh1 = (const float*)d_in[7];
  const float* Wh2 = (const float*)d_in[8];
  const float* bh2 = (const float*)d_in[9];

  const int N = in_sizes[0] / 256;
  const int E = in_sizes[1] / 2;
  const int* src = ei;
  const int* dst = ei + E;

  // Workspace layout (floats): dinv[N] | bufA[N*64] | bufB[N*64]
  float* dinv = (float*)d_ws;
  float* bufA = dinv + N;
  float* bufB = bufA + (long long)N * 64;

  const int TB = 256;
  dim3 blk(TB);

  // 1) degree + normalization
  deg_init_kernel<<<(N + TB - 1) / TB, blk, 0, stream>>>(dinv, N);
  deg_accum_kernel<<<(E + TB - 1) / TB, blk, 0, stream>>>(dst, dinv, E);
  deg_to_dinv_kernel<<<(N + TB - 1) / TB, blk, 0, stream>>>(dinv, N);

  // GEMM grid: one wave per 16 rows, 8 waves per block
  long long waves = (N + 15) / 16;
  int gemm_blocks = (int)((waves + 7) / 8);

  long long nodeVec = (long long)N * 16;              // float4 lanes
  int nv_blocks = (int)((nodeVec + TB - 1) / TB);
  long long edgeT = (long long)E * 32;                // one wave per edge
  int ea_blocks = (int)((edgeT + TB - 1) / TB);

  // 2) layer 1: h = relu(Aggregate(x @ W1) + b1)
  gemm64_wmma_f32_kernel<<<gemm_blocks, blk, 256 * 64 * 4, stream>>>(x, W1, bufA, N, 256);
  self_loop_init_kernel<<<nv_blocks, blk, 0, stream>>>(bufA, dinv, bufB, N);
  edge_agg_kernel<<<ea_blocks, blk, 0, stream>>>(src, dst, dinv, bufA, bufB, E);
  bias_relu_kernel<<<nv_blocks, blk, 0, stream>>>(bufB, b1, N);

  // 3) layer 2: h = relu(Aggregate(h @ W2) + b2)
  gemm64_wmma_f32_kernel<<<gemm_blocks, blk, 64 * 64 * 4, stream>>>(bufB, W2, bufA, N, 64);
  self_loop_init_kernel<<<nv_blocks, blk, 0, stream>>>(bufA, dinv, bufB, N);
  edge_agg_kernel<<<ea_blocks, blk, 0, stream>>>(src, dst, dinv, bufA, bufB, E);
  bias_relu_kernel<<<nv_blocks, blk, 0, stream>>>(bufB, b2, N);

  // 4) MLP head -> d_out [N, 2]
  head_kernel<<<(N + 127) / 128, dim3(128), 0, stream>>>(
      bufB, Wh1, bh1, Wh2, bh2, (float*)d_out, N);
}